// decoder_23270132810397
// MI455X (gfx1250) — compile-verified
//
#include <hip/hip_runtime.h>

// GRU decoder: 512-step recurrence, batch 8192, HID 10, IN 64.
// One wave = 16 batch rows (M dim of WMMA tiles). All GEMMs via
// V_WMMA_F32_16X16X4_F32 (fp32, matches reference precision).
// Per-gate N-tiles keep gating lane-aligned; C->A layout conversion via
// wave-private LDS with conflict-free strides.

typedef float v2f __attribute__((ext_vector_type(2)));
typedef float v8f __attribute__((ext_vector_type(8)));

#define BATCH   8192
#define HID     10
#define IN_DIM  64
#define XSTRIDE 68   // 16x64 x-tile row stride (conflict-free, 8B aligned)
#define HSTRIDE 20   // 16x16 h/o-tile row stride (conflict-free, 8B aligned)
#define WAVES   4    // waves per block
#define TPB     (WAVES * 32)

static __device__ inline v8f wmma4(v2f a, v2f b, v8f c) {
  return __builtin_amdgcn_wmma_f32_16x16x4_f32(
      /*neg_a=*/false, a, /*neg_b=*/false, b,
      /*c_mod=*/(short)0, c, /*reuse_a=*/false, /*reuse_b=*/false);
}

static __device__ inline float sigmoidf_(float x) {
  return 1.0f / (1.0f + __expf(-x));
}
static __device__ inline float tanhf_(float x) {
  x = fminf(fmaxf(x, -15.0f), 15.0f);
  float e = __expf(-2.0f * x);
  return (1.0f - e) / (1.0f + e);
}

__global__ __launch_bounds__(TPB)
void gru_decoder_kernel(const float* __restrict__ hidden,
                        const float* __restrict__ w_ih,
                        const float* __restrict__ w_hh,
                        const float* __restrict__ b_ih,
                        const float* __restrict__ b_hh,
                        const float* __restrict__ l1_w,
                        const float* __restrict__ l1_b,
                        const float* __restrict__ l2_w,
                        const float* __restrict__ l2_b,
                        const int*   __restrict__ step_p,
                        float*       __restrict__ out)
{
  const int tid  = threadIdx.x;
  const int lane = tid & 31;
  const int wave = tid >> 5;
  const int half = lane >> 4;   // 0: K pair {0,1} / M 0-7 ; 1: K pair {2,3} / M 8-15
  const int ln   = lane & 15;   // N column (B/C frags) or M row (A frags)
  const int b0   = (blockIdx.x * WAVES + wave) * 16;
  const int steps = step_p[0];

  // w_ih B-fragments, per gate g, K-chunk c, lane l: float2 = W^T[k..k+1][n]
  __shared__ __align__(16) float wf[3 * 16 * 32 * 2];          // 12 KB
  __shared__ __align__(16) float xbuf[WAVES][16 * XSTRIDE];    // x state tiles
  __shared__ __align__(16) float hbuf[WAVES][16 * HSTRIDE];    // h state tiles
  __shared__ __align__(16) float obuf[WAVES][16 * HSTRIDE];    // o tiles

  // ---- stage w_ih as WMMA B fragments in LDS (block-shared) ----
  for (int e = tid; e < 3 * 16 * 32; e += TPB) {
    int l  = e & 31;
    int c  = (e >> 5) & 15;
    int g  = e >> 9;
    int eh = l >> 4, en = l & 15;
    int k  = 4 * c + 2 * eh;
    float v0 = 0.f, v1 = 0.f;
    if (en < HID) {
      const float* row = w_ih + (size_t)(g * HID + en) * IN_DIM;
      v0 = row[k]; v1 = row[k + 1];
    }
    wf[e * 2 + 0] = v0;
    wf[e * 2 + 1] = v1;
  }

  // ---- small weights as register-resident B fragments ----
  v2f Bh[3][3];            // w_hh per gate, 3 K-chunks (K=10 padded to 12)
  for (int g = 0; g < 3; ++g)
    for (int c = 0; c < 3; ++c) {
      int k = 4 * c + 2 * half;
      float v0 = 0.f, v1 = 0.f;
      if (ln < HID) {
        const float* row = w_hh + (size_t)(g * HID + ln) * HID;
        if (k     < HID) v0 = row[k];
        if (k + 1 < HID) v1 = row[k + 1];
      }
      Bh[g][c] = v2f{v0, v1};
    }
  v2f Bl1[3];
  for (int c = 0; c < 3; ++c) {
    int k = 4 * c + 2 * half;
    float v0 = 0.f, v1 = 0.f;
    if (ln < HID) {
      const float* row = l1_w + (size_t)ln * HID;
      if (k     < HID) v0 = row[k];
      if (k + 1 < HID) v1 = row[k + 1];
    }
    Bl1[c] = v2f{v0, v1};
  }
  v2f Bl2[3][4];           // l2_w^T: K=10 (3 chunks) x N=64 (4 tiles)
  for (int c = 0; c < 3; ++c)
    for (int nt = 0; nt < 4; ++nt) {
      int k = 4 * c + 2 * half;
      int n = nt * 16 + ln;
      const float* row = l2_w + (size_t)n * HID;
      float v0 = (k     < HID) ? row[k]     : 0.f;
      float v1 = (k + 1 < HID) ? row[k + 1] : 0.f;
      Bl2[c][nt] = v2f{v0, v1};
    }

  // per-lane biases (zero in pad columns so pads stay exactly 0)
  float br  = (ln < HID) ? (b_ih[ln]           + b_hh[ln])           : 0.f;
  float bz  = (ln < HID) ? (b_ih[HID + ln]     + b_hh[HID + ln])     : 0.f;
  float bni = (ln < HID) ?  b_ih[2 * HID + ln]                        : 0.f;
  float bnh = (ln < HID) ?  b_hh[2 * HID + ln]                        : 0.f;
  float bo  = (ln < HID) ?  l1_b[ln]                                  : 0.f;
  float bx[4];
  for (int nt = 0; nt < 4; ++nt) bx[nt] = l2_b[nt * 16 + ln];

  // ---- initial state ----
  float* xw = xbuf[wave];
  float* hw = hbuf[wave];
  float* ow = obuf[wave];
  for (int i = lane; i < 16 * XSTRIDE; i += 32) xw[i] = 0.f;   // x0 = 0

  v8f hC;   // previous h as C-fragment (needed for blend), pads = 0
  #pragma unroll
  for (int r = 0; r < 8; ++r) {
    int M = r + 8 * half;
    hC[r] = (ln < HID) ? hidden[(size_t)(b0 + M) * HID + ln] : 0.f;
  }
  #pragma unroll
  for (int r = 0; r < 8; ++r) hw[(r + 8 * half) * HSTRIDE + ln] = hC[r];

  __syncthreads();   // wf ready for all waves

  const size_t outrow = (size_t)steps * HID;

  #pragma unroll 1
  for (int t = 0; t < steps; ++t) {
    // ---- gi (+gh for r,z): accumulate x-part then h-part ----
    v8f aR = {}, aZ = {}, aNi = {}, aNh = {};
    #pragma unroll
    for (int c = 0; c < 16; ++c) {
      const float* px = xw + ln * XSTRIDE + 4 * c + 2 * half;
      v2f xa = {px[0], px[1]};
      const float* p0 = wf + ((0 * 16 + c) * 32 + lane) * 2;
      const float* p1 = wf + ((1 * 16 + c) * 32 + lane) * 2;
      const float* p2 = wf + ((2 * 16 + c) * 32 + lane) * 2;
      v2f wr = {p0[0], p0[1]};
      v2f wz = {p1[0], p1[1]};
      v2f wn = {p2[0], p2[1]};
      aR  = wmma4(xa, wr, aR);
      aZ  = wmma4(xa, wz, aZ);
      aNi = wmma4(xa, wn, aNi);
    }
    #pragma unroll
    for (int c = 0; c < 3; ++c) {
      const float* ph = hw + ln * HSTRIDE + 4 * c + 2 * half;
      v2f ha = {ph[0], ph[1]};
      aR  = wmma4(ha, Bh[0][c], aR);
      aZ  = wmma4(ha, Bh[1][c], aZ);
      aNh = wmma4(ha, Bh[2][c], aNh);
    }

    // ---- GRU gating (lane-aligned, pure VALU) ----
    v8f hNew;
    #pragma unroll
    for (int r = 0; r < 8; ++r) {
      float rg = sigmoidf_(aR[r] + br);
      float zg = sigmoidf_(aZ[r] + bz);
      float ng = tanhf_(aNi[r] + bni + rg * (aNh[r] + bnh));
      hNew[r] = (1.0f - zg) * ng + zg * hC[r];
    }
    hC = hNew;
    #pragma unroll
    for (int r = 0; r < 8; ++r) hw[(r + 8 * half) * HSTRIDE + ln] = hNew[r];

    // ---- o = h_new @ l1^T + l1_b ----
    v8f aO = {};
    #pragma unroll
    for (int c = 0; c < 3; ++c) {
      const float* ph = hw + ln * HSTRIDE + 4 * c + 2 * half;
      v2f ha = {ph[0], ph[1]};
      aO = wmma4(ha, Bl1[c], aO);
    }
    #pragma unroll
    for (int r = 0; r < 8; ++r) aO[r] += bo;

    // emit o at reversed time index
    const size_t tcol = (size_t)(steps - 1 - t) * HID;
    if (ln < HID) {
      #pragma unroll
      for (int r = 0; r < 8; ++r) {
        int M = r + 8 * half;
        out[(size_t)(b0 + M) * outrow + tcol + ln] = aO[r];
      }
    }
    #pragma unroll
    for (int r = 0; r < 8; ++r) ow[(r + 8 * half) * HSTRIDE + ln] = aO[r];

    // ---- x_next = o @ l2^T + l2_b (4 N-tiles of 16) ----
    #pragma unroll
    for (int nt = 0; nt < 4; ++nt) {
      v8f aX = {};
      #pragma unroll
      for (int c = 0; c < 3; ++c) {
        const float* po = ow + ln * HSTRIDE + 4 * c + 2 * half;
        v2f oa = {po[0], po[1]};
        aX = wmma4(oa, Bl2[c][nt], aX);
      }
      #pragma unroll
      for (int r = 0; r < 8; ++r)
        xw[(r + 8 * half) * XSTRIDE + nt * 16 + ln] = aX[r] + bx[nt];
    }
  }
}

extern "C" void kernel_launch(void* const* d_in, const int* in_sizes, int n_in,
                              void* d_out, int out_size, void* d_ws, size_t ws_size,
                              hipStream_t stream) {
  const float* hidden = (const float*)d_in[0];
  const float* w_ih   = (const float*)d_in[1];
  const float* w_hh   = (const float*)d_in[2];
  const float* b_ih   = (const float*)d_in[3];
  const float* b_hh   = (const float*)d_in[4];
  const float* l1_w   = (const float*)d_in[5];
  const float* l1_b   = (const float*)d_in[6];
  const float* l2_w   = (const float*)d_in[7];
  const float* l2_b   = (const float*)d_in[8];
  const int*   step   = (const int*)d_in[9];
  float* out = (float*)d_out;

  dim3 grid(BATCH / (WAVES * 16));   // 128 blocks x 4 waves x 16 rows = 8192
  dim3 block(TPB);
  gru_decoder_kernel<<<grid, block, 0, stream>>>(hidden, w_ih, w_hh, b_ih, b_hh,
                                                 l1_w, l1_b, l2_w, l2_b, step, out);
}